// AllAtomFAPE_73718818668725
// MI455X (gfx1250) — compile-verified
//
#include <hip/hip_runtime.h>

typedef float v2f __attribute__((ext_vector_type(2)));
typedef float v8f __attribute__((ext_vector_type(8)));

#define NFRAMES 2048   // 256 res * 8 frames
#define NATOMS  3584   // 256 res * 14 atoms
#define NRES    256
#define NAT     14
#define FTILES  (NFRAMES / 16)   // 128
#define ATILES  (NATOMS  / 16)   // 224

__global__ __launch_bounds__(4) void fape_init(float* acc) {
    if (threadIdx.x < 4) acc[threadIdx.x] = 0.0f;
}

__global__ __launch_bounds__(32)
void fape_main(const float* __restrict__ Rp, const float* __restrict__ tp,
               const float* __restrict__ ppos, const float* __restrict__ amask,
               const float* __restrict__ Rt, const float* __restrict__ tt,
               const float* __restrict__ tposa, const float* __restrict__ tamask,
               const float* __restrict__ smask, const float* __restrict__ fmask,
               float* __restrict__ acc)
{
    const int b    = blockIdx.y;
    const int ft   = blockIdx.x;
    const int lane = threadIdx.x;
    const int col  = lane & 15;
    const bool hi  = lane >= 16;      // lanes 16-31 carry K=2,3 of A/B operands

    // ---------- Build the six 16x4 f32 A operands (held for the whole kernel) ----------
    // A_pred_i row m : [ R[0][i], R[1][i], R[2][i], -(R^T t)_i ]   (homogeneous affine)
    // A_true_i row m : [-R[0][i],-R[1][i],-R[2][i], +(R^T t)_i ]   (negation folded in)
    const int f = ft * 16 + col;
    const float* rp  = Rp + ((size_t)b * NFRAMES + f) * 9;
    const float* rt  = Rt + ((size_t)b * NFRAMES + f) * 9;
    const float* tpv = tp + ((size_t)b * NFRAMES + f) * 3;
    const float* ttv = tt + ((size_t)b * NFRAMES + f) * 3;
    const float tp0 = tpv[0], tp1 = tpv[1], tp2 = tpv[2];
    const float tt0 = ttv[0], tt1 = ttv[1], tt2 = ttv[2];

    v2f aP[3], aT[3];
#pragma unroll
    for (int i = 0; i < 3; ++i) {
        float rp0 = rp[0*3+i], rp1 = rp[1*3+i], rp2 = rp[2*3+i];
        float rt0 = rt[0*3+i], rt1 = rt[1*3+i], rt2 = rt[2*3+i];
        float tinvP  = -(rp0*tp0 + rp1*tp1 + rp2*tp2);
        float tinvTn =  (rt0*tt0 + rt1*tt1 + rt2*tt2);
        aP[i].x = hi ?  rp2   :  rp0;
        aP[i].y = hi ?  tinvP :  rp1;
        aT[i].x = hi ? -rt2   : -rt0;
        aT[i].y = hi ?  tinvTn : -rt1;
    }

    // frame-mask per C register: row M = r + 8*hi, frame = ft*16 + M
    float fmv[8];
#pragma unroll
    for (int r = 0; r < 8; ++r) {
        int fidx = ft * 16 + r + (hi ? 8 : 0);
        fmv[r] = smask[b * NRES + (fidx >> 3)] * fmask[b * NFRAMES + fidx];
    }

    float num = 0.0f, den = 0.0f;

    for (int at = 0; at < ATILES; ++at) {
        // ---------- B operands: 4x16, column n = [pos_x, pos_y, pos_z, 1] ----------
        const int aidx = at * 16 + col;
        const size_t ao = (size_t)b * NATOMS + aidx;
        const float* pp = ppos  + ao * 3;
        const float* pt = tposa + ao * 3;
        v2f bP, bT;
        bP.x = hi ? pp[2] : pp[0];
        bP.y = hi ? 1.0f  : pp[1];
        bT.x = hi ? pt[2] : pt[0];
        bT.y = hi ? 1.0f  : pt[1];
        const float am = amask[ao] * tamask[ao] * smask[b * NRES + aidx / NAT];

        // ---------- 6 WMMAs: diff_i[f,a] accumulated across pred + (-true) ----------
        v8f c0 = {}, c1 = {}, c2 = {};
        c0 = __builtin_amdgcn_wmma_f32_16x16x4_f32(false, aP[0], false, bP, (short)0, c0, false, false);
        c0 = __builtin_amdgcn_wmma_f32_16x16x4_f32(false, aT[0], false, bT, (short)0, c0, false, false);
        c1 = __builtin_amdgcn_wmma_f32_16x16x4_f32(false, aP[1], false, bP, (short)0, c1, false, false);
        c1 = __builtin_amdgcn_wmma_f32_16x16x4_f32(false, aT[1], false, bT, (short)0, c1, false, false);
        c2 = __builtin_amdgcn_wmma_f32_16x16x4_f32(false, aP[2], false, bP, (short)0, c2, false, false);
        c2 = __builtin_amdgcn_wmma_f32_16x16x4_f32(false, aT[2], false, bT, (short)0, c2, false, false);

        // ---------- err = min(sqrt(|d|^2 + eps), 10), masked accumulate ----------
#pragma unroll
        for (int r = 0; r < 8; ++r) {
            float dx = c0[r], dy = c1[r], dz = c2[r];
            float e = __builtin_sqrtf(dx*dx + dy*dy + dz*dz + 1e-4f);
            e = fminf(e, 10.0f);
            float w = fmv[r] * am;
            num += e * w;
            den += w;
        }
    }

    // wave32 butterfly reduction
#pragma unroll
    for (int off = 16; off >= 1; off >>= 1) {
        num += __shfl_xor(num, off, 32);
        den += __shfl_xor(den, off, 32);
    }
    if (lane == 0) {
        atomicAdd(&acc[b * 2 + 0], num);
        atomicAdd(&acc[b * 2 + 1], den);
    }
}

__global__ __launch_bounds__(2) void fape_final(const float* __restrict__ acc,
                                                float* __restrict__ out) {
    int b = threadIdx.x;
    if (b < 2) {
        float n = acc[b * 2 + 0];
        float d = acc[b * 2 + 1];
        out[b] = n / (fmaxf(d, 1.0f) * 10.0f);
    }
}

extern "C" void kernel_launch(void* const* d_in, const int* in_sizes, int n_in,
                              void* d_out, int out_size, void* d_ws, size_t ws_size,
                              hipStream_t stream) {
    const float* Rp     = (const float*)d_in[0];  // predicted_frames_R      [2,256,8,3,3]
    const float* tp     = (const float*)d_in[1];  // predicted_frames_t      [2,256,8,3]
    const float* ppos   = (const float*)d_in[2];  // predicted_atom_positions[2,256,14,3]
    const float* amask  = (const float*)d_in[3];  // atom_mask               [2,256,14]
    const float* Rt     = (const float*)d_in[4];  // true_frames_R
    const float* tt     = (const float*)d_in[5];  // true_frames_t
    const float* tposa  = (const float*)d_in[6];  // true_atom_positions
    const float* tamask = (const float*)d_in[7];  // true_atom_mask
    const float* smask  = (const float*)d_in[8];  // seq_mask                [2,256]
    const float* fmask  = (const float*)d_in[9];  // frame_mask              [2,256,8]
    float* out = (float*)d_out;
    float* acc = (float*)d_ws;   // 4 floats: {num,den} per batch

    fape_init<<<1, 4, 0, stream>>>(acc);
    fape_main<<<dim3(FTILES, 2), 32, 0, stream>>>(Rp, tp, ppos, amask,
                                                  Rt, tt, tposa, tamask,
                                                  smask, fmask, acc);
    fape_final<<<1, 2, 0, stream>>>(acc, out);
}